// ExpertRouter_85504208929566
// MI455X (gfx1250) — compile-verified
//
#include <hip/hip_runtime.h>
#include <math.h>

typedef __attribute__((ext_vector_type(2))) float v2f;
typedef __attribute__((ext_vector_type(4))) float v4f;
typedef __attribute__((ext_vector_type(8))) float v8f;

#define EXPERTS   32
#define TOPK      4
#define HID       2048
#define TOKENS    16384            // 4 * 4096
#define TILES     (TOKENS / 16)    // 1024 wave-tiles (16 tokens each)
#define NWAVES    4                // 128 threads / wave32
#define BLOCKT    (NWAVES * 32)    // 128 threads per block
#define NBLOCKS   256              // 256 blocks x 4 waves = 1024 waves = 1 tile each
#define WSTRIDE   2052             // weight LDS row stride; 2052 % 64 == 4 -> all 64 banks hit, 0 conflicts
#define PSTRIDE   36               // per-wave prob-row stride; t*36 mod 64 distinct for t=0..15
#define WLDS      (EXPERTS * WSTRIDE)       // 65664 floats
#define PLDS      (NWAVES * 16 * PSTRIDE)   // 2304 floats
#define LDS_FLOATS (WLDS + PLDS + NWAVES * EXPERTS)  // + wave partials: 68096 floats = 272384 B

__global__ __launch_bounds__(BLOCKT)
void router_main_kernel(const float* __restrict__ H,     // [16384][2048]
                        const float* __restrict__ W,     // [32][2048]
                        const float* __restrict__ bias,  // [32]
                        float* __restrict__ out,         // mask | loss | indices
                        float* __restrict__ wsPartials)  // [NBLOCKS][32]
{
    extern __shared__ float smem[];
    float* sW = smem;

    const int tid  = threadIdx.x;
    const int wave = tid >> 5;
    const int lane = tid & 31;

    // ---- stage router weights into LDS (padded rows), float4 copies ----
    for (int idx = tid; idx < EXPERTS * (HID / 4); idx += blockDim.x) {
        const int row = idx >> 9;      // / 512
        const int c4  = idx & 511;
        v4f v = *(const v4f*)(W + row * HID + c4 * 4);
        *(v4f*)(sW + row * WSTRIDE + c4 * 4) = v;
    }
    __syncthreads();

    // per-lane WMMA operand addressing (ISA: 32-bit A 16x4 / B 4x16 layouts)
    const int n    = lane & 15;            // M row (A) / N col (B) within tile
    const int koff = (lane >> 4) << 1;     // lanes 16-31 hold K+2,K+3
    const float* bW0 = sW + n * WSTRIDE + koff;          // experts 0-15
    const float* bW1 = sW + (n + 16) * WSTRIDE + koff;   // experts 16-31
    float* sP = smem + WLDS + wave * 16 * PSTRIDE;       // per-wave [16][PSTRIDE]

    const float bias0 = bias[n];
    const float bias1 = bias[n + 16];

    float pl = 0.f;  // per-lane expert-load partial (expert = lane)

    float* dMask = out;                          // [16384][32]
    float* dIdx  = out + TOKENS * EXPERTS + 1;   // after mask + scalar loss

    for (int tile = blockIdx.x * NWAVES + wave; tile < TILES; tile += gridDim.x * NWAVES) {
        const int tokenBase = tile * 16;
        const float* aPtr = H + (size_t)(tokenBase + n) * HID + koff;

        v8f acc0 = {0.f,0.f,0.f,0.f,0.f,0.f,0.f,0.f};
        v8f acc1 = {0.f,0.f,0.f,0.f,0.f,0.f,0.f,0.f};

        #pragma unroll 8
        for (int k = 0; k < HID; k += 4) {
            v2f a  = *(const v2f*)(aPtr + k);
            v2f b0 = *(const v2f*)(bW0 + k);
            v2f b1 = *(const v2f*)(bW1 + k);
            acc0 = __builtin_amdgcn_wmma_f32_16x16x4_f32(false, a, false, b0,
                                                         (short)0, acc0, false, false);
            acc1 = __builtin_amdgcn_wmma_f32_16x16x4_f32(false, a, false, b1,
                                                         (short)0, acc1, false, false);
        }

        // ---- scatter logits (+bias) into per-wave LDS rows ----
        // C/D layout: VGPR i <-> M=i (lanes 0-15) / M=i+8 (lanes 16-31), N=lane&15
        const int rowAdd = (lane >> 4) << 3;
        #pragma unroll
        for (int i = 0; i < 8; ++i) {
            sP[(i + rowAdd) * PSTRIDE + n]      = acc0[i] + bias0;
            sP[(i + rowAdd) * PSTRIDE + n + 16] = acc1[i] + bias1;
        }

        int   i0 = 0, i1 = 0, i2 = 0, i3 = 0;
        float p0 = 0.f, p1 = 0.f, p2 = 0.f, p3 = 0.f;

        if (lane < 16) {   // lane t owns token tokenBase+t
            float* row = sP + lane * PSTRIDE;
            // softmax over 32 experts
            float m = row[0];
            #pragma unroll
            for (int e = 1; e < EXPERTS; ++e) m = fmaxf(m, row[e]);
            float s = 0.f;
            #pragma unroll
            for (int e = 0; e < EXPERTS; ++e) { float p = __expf(row[e] - m); row[e] = p; s += p; }
            const float inv = 1.f / s;
            #pragma unroll
            for (int e = 0; e < EXPERTS; ++e) row[e] *= inv;

            // top-4, strict '>' => first occurrence wins ties (matches lax.top_k)
            float best = -1.f;
            #pragma unroll
            for (int e = 0; e < EXPERTS; ++e) { float v = row[e]; if (v > best) { best = v; i0 = e; } }
            p0 = best; best = -1.f;
            #pragma unroll
            for (int e = 0; e < EXPERTS; ++e) { if (e == i0) continue; float v = row[e]; if (v > best) { best = v; i1 = e; } }
            p1 = best; best = -1.f;
            #pragma unroll
            for (int e = 0; e < EXPERTS; ++e) { if (e == i0 || e == i1) continue; float v = row[e]; if (v > best) { best = v; i2 = e; } }
            p2 = best; best = -1.f;
            #pragma unroll
            for (int e = 0; e < EXPERTS; ++e) { if (e == i0 || e == i1 || e == i2) continue; float v = row[e]; if (v > best) { best = v; i3 = e; } }
            p3 = best;
        }

        // ---- all lanes: expert-load column sums (deterministic order) ----
        {
            float csum = 0.f;
            #pragma unroll
            for (int t = 0; t < 16; ++t) csum += sP[t * PSTRIDE + lane];
            pl += csum;
        }

        // ---- overwrite prob rows with dispatch-mask rows, write indices ----
        if (lane < 16) {
            float* row = sP + lane * PSTRIDE;
            #pragma unroll
            for (int e = 0; e < EXPERTS; ++e) row[e] = 0.f;
            const float invt = 1.f / (p0 + p1 + p2 + p3);
            row[i0] = p0 * invt; row[i1] = p1 * invt;
            row[i2] = p2 * invt; row[i3] = p3 * invt;
            float* ip = dIdx + (size_t)(tokenBase + lane) * TOPK;
            ip[0] = (float)i0; ip[1] = (float)i1; ip[2] = (float)i2; ip[3] = (float)i3;
        }

        // ---- coalesced 128B dispatch-mask row stores ----
        #pragma unroll
        for (int t = 0; t < 16; ++t)
            dMask[(size_t)(tokenBase + t) * EXPERTS + lane] = sP[t * PSTRIDE + lane];
    }

    // ---- deterministic block-level expert-load reduction ----
    float* wavePart = smem + WLDS + PLDS;
    wavePart[wave * EXPERTS + lane] = pl;
    __syncthreads();
    if (tid < EXPERTS) {
        float s = 0.f;
        #pragma unroll
        for (int w = 0; w < NWAVES; ++w) s += wavePart[w * EXPERTS + tid];
        wsPartials[blockIdx.x * EXPERTS + tid] = s;
    }
}

__global__ void router_finalize_kernel(const float* __restrict__ wsPartials,
                                       float* __restrict__ out)
{
    if (threadIdx.x == 0 && blockIdx.x == 0) {
        const float target = 1.0f / (float)EXPERTS;
        const float logt   = logf(target);
        float loss = 0.f;
        for (int e = 0; e < EXPERTS; ++e) {
            float s = 0.f;
            for (int b = 0; b < NBLOCKS; ++b) s += wsPartials[b * EXPERTS + e];
            const float load = s / (float)TOKENS;
            loss += target * (logt - logf(load));
        }
        out[TOKENS * EXPERTS] = loss * (1.0f / (float)EXPERTS) * 1e-4f;
    }
}

extern "C" void kernel_launch(void* const* d_in, const int* in_sizes, int n_in,
                              void* d_out, int out_size, void* d_ws, size_t ws_size,
                              hipStream_t stream) {
    const float* H    = (const float*)d_in[0];   // [4,4096,2048] f32
    const float* W    = (const float*)d_in[1];   // [32,2048] f32
    const float* bias = (const float*)d_in[2];   // [32] f32
    float* out        = (float*)d_out;
    float* partials   = (float*)d_ws;            // NBLOCKS*32 floats

    const size_t shmem = (size_t)LDS_FLOATS * sizeof(float);  // 272384 B of WGP LDS
    router_main_kernel<<<NBLOCKS, BLOCKT, shmem, stream>>>(H, W, bias, out, partials);
    router_finalize_kernel<<<1, 32, 0, stream>>>(partials, out);
}